// InvariantGCN_62672162783236
// MI455X (gfx1250) — compile-verified
//
#include <hip/hip_runtime.h>
#include <hip/hip_bf16.h>
#include <math.h>

// ---------------------------------------------------------------------------
// InvariantGCN forward for MI455X (gfx1250, wave32, WMMA).
// Sizes fixed by the reference: V=1000 N=4096 E=32768 D=256 H=512 O=256 A=128
// L=64 B=8.  All GEMMs run on V_WMMA_F32_16X16X4_F32 (fp32-exact vs reference).
// ---------------------------------------------------------------------------

#define V_  1000
#define N_  4096
#define E_  32768
#define D_  256
#define H_  512
#define O_  256
#define A_  128
#define L_  64
#define B_  8

typedef float v2f __attribute__((ext_vector_type(2)));
typedef float v8f __attribute__((ext_vector_type(8)));

// ---------------------------------------------------------------------------
// Generic fp32 WMMA GEMM: C[M,N] = op(A[M,K] @ B) + bias, one wave per 16x16
// tile.  BT=1 means memory holds Bt[N,K] (i.e. B = Bt^T, used for W^T GEMMs).
// ---------------------------------------------------------------------------
template<int BT, int RELU>
__global__ void gemm_wmma_f32(const float* __restrict__ A,
                              const float* __restrict__ Bm,
                              float* __restrict__ C,
                              const float* __restrict__ bias,
                              int M, int K, int N) {
  const int lane = threadIdx.x & 31;
  const int wave = threadIdx.x >> 5;            // blockDim.x == 128 -> 4 waves
  const int mt = (M + 15) >> 4;
  const int nt = N >> 4;
  const int tile = blockIdx.x * 4 + wave;       // wave-uniform
  if (tile >= mt * nt) return;
  const int tm = tile / nt;
  const int tn = tile - tm * nt;
  const int l15  = lane & 15;
  const int half = lane >> 4;                   // 0: K+{0,1}, 1: K+{2,3}
  const int arow = tm * 16 + l15;
  const int bcol = tn * 16 + l15;
  const bool aok = (arow < M);

  const float* __restrict__ Arow = A + (size_t)arow * K;
  const float* __restrict__ Bcol = BT ? (Bm + (size_t)bcol * K) : Bm;

  v8f acc = {0.f, 0.f, 0.f, 0.f, 0.f, 0.f, 0.f, 0.f};
  for (int k0 = 0; k0 < K; k0 += 4) {
    const int k = k0 + half * 2;
    __builtin_prefetch(Arow + k0 + 32, 0, 0);   // global_prefetch_b8
    v2f a, b;
    a.x = aok ? Arow[k]     : 0.f;              // select, keeps EXEC all-ones
    a.y = aok ? Arow[k + 1] : 0.f;
    if (BT) {
      b.x = Bcol[k];
      b.y = Bcol[k + 1];
    } else {
      b.x = Bm[(size_t)k       * N + bcol];
      b.y = Bm[(size_t)(k + 1) * N + bcol];
    }
    acc = __builtin_amdgcn_wmma_f32_16x16x4_f32(false, a, false, b,
                                                (short)0, acc, false, false);
  }

  const float bv = bias ? bias[bcol] : 0.f;
  const int r0 = tm * 16 + half * 8;
#pragma unroll
  for (int i = 0; i < 8; ++i) {
    const int r = r0 + i;
    if (r < M) {
      float v = acc[i] + bv;
      if (RELU) v = fmaxf(v, 0.f);
      C[(size_t)r * N + bcol] = v;
    }
  }
}

// ---------------------------------------------------------------------------
// Utility fills
// ---------------------------------------------------------------------------
__global__ void fill_f32(float* __restrict__ p, size_t n, float v) {
  size_t i = (size_t)blockIdx.x * blockDim.x + threadIdx.x;
  if (i < n) p[i] = v;
}

// deg[b][dst] += 1 over edges (deg pre-filled with 1.0 for the self loop)
__global__ void edge_deg(const int* __restrict__ ei, float* __restrict__ deg) {
  int i = blockIdx.x * blockDim.x + threadIdx.x;        // b*E + e
  if (i >= B_ * E_) return;
  int b = i >> 15;                                      // E_ == 1<<15
  int e = i & (E_ - 1);
  int dst = ei[(size_t)b * 2 * E_ + E_ + e];
  atomicAdd(&deg[b * N_ + dst], 1.0f);
}

__global__ void deg_to_dinv(const float* __restrict__ deg,
                            float* __restrict__ dinv) {
  int i = blockIdx.x * blockDim.x + threadIdx.x;
  if (i < B_ * N_) dinv[i] = rsqrtf(deg[i]);
}

// agg[b][dst][:] += table_xw[tok[b][src]][:] * dinv[src]*dinv[dst]
// one block (256 threads == D) per edge
__global__ void edge_agg(const int* __restrict__ ei,
                         const int* __restrict__ tok,
                         const float* __restrict__ txw,
                         const float* __restrict__ dinv,
                         float* __restrict__ agg) {
  int idx = blockIdx.x;                                 // b*E + e
  int b = idx >> 15;
  int e = idx & (E_ - 1);
  const int* eb = ei + (size_t)b * 2 * E_;
  int src = eb[e];
  int dst = eb[E_ + e];
  float coef = dinv[b * N_ + src] * dinv[b * N_ + dst];
  int tk = tok[b * N_ + src];
  int d = threadIdx.x;
  atomicAdd(&agg[((size_t)b * N_ + dst) * D_ + d],
            txw[(size_t)tk * D_ + d] * coef);
}

// graph_repr = relu(agg + table_xw[tok]*dinv^2 + gcn_b)   (in place over agg)
__global__ void graph_finish(const int* __restrict__ tok,
                             const float* __restrict__ txw,
                             const float* __restrict__ dinv,
                             const float* __restrict__ gcn_b,
                             float* __restrict__ agg) {
  int idx = blockIdx.x;                                 // b*N + n
  int d = threadIdx.x;
  int tk = tok[idx];
  float di = dinv[idx];
  size_t o = (size_t)idx * D_ + d;
  float v = agg[o] + txw[(size_t)tk * D_ + d] * (di * di) + gcn_b[d];
  agg[o] = fmaxf(v, 0.f);
}

// out[b*T+t][:] = emb[tok<V ? tok : 0] + graph[b][tok<V ? 0 : tok-V+1]
__global__ void mixed_embed(const int* __restrict__ seq,
                            const float* __restrict__ emb,
                            const float* __restrict__ graph,
                            float* __restrict__ out, int T) {
  int idx = blockIdx.x;                                 // b*T + t
  int b = idx / T;
  int d = threadIdx.x;
  int tk = seq[idx];
  int eidx = (tk < V_) ? tk : 0;
  int gidx = (tk < V_) ? 0 : (tk - V_ + 1);
  out[(size_t)idx * D_ + d] =
      emb[(size_t)eidx * D_ + d] +
      graph[((size_t)b * N_ + gidx) * D_ + d];
}

// ---------------------------------------------------------------------------
// Persistent GRU: one workgroup, h kept in LDS for all 64 steps.
// Per step: gh[8,768] = h @ Whh^T (WMMA, A from LDS) + b_hh, then gate math.
// 512 threads = 16 waves; 48 N-tiles -> 3 tiles per wave.
// ---------------------------------------------------------------------------
__global__ void gru_kernel(const float* __restrict__ xp,   // [B,L,3O] (b_ih added)
                           const float* __restrict__ Whh,  // [3O,O]
                           const float* __restrict__ bhh,  // [3O]
                           float* __restrict__ inv) {      // [B,O]
  __shared__ float h[16 * O_];        // rows 8..15 stay zero (M padding)
  __shared__ float gh[B_ * 3 * O_];
  const int tid  = threadIdx.x;       // 512 threads
  const int lane = tid & 31;
  const int wave = tid >> 5;
  const int l15  = lane & 15;
  const int half = lane >> 4;

  for (int i = tid; i < 16 * O_; i += 512) h[i] = 0.f;
  __syncthreads();

  for (int t = 0; t < L_; ++t) {
    // --- gh = h @ Whh^T + bhh -------------------------------------------
    for (int tt = wave; tt < (3 * O_) / 16; tt += 16) {
      const int gcol = tt * 16 + l15;
      const float* __restrict__ Bc = Whh + (size_t)gcol * O_;
      v8f acc = {0.f, 0.f, 0.f, 0.f, 0.f, 0.f, 0.f, 0.f};
      for (int k0 = 0; k0 < O_; k0 += 4) {
        const int k = k0 + half * 2;
        v2f a, b;
        a.x = h[l15 * O_ + k];
        a.y = h[l15 * O_ + k + 1];
        b.x = Bc[k];
        b.y = Bc[k + 1];
        acc = __builtin_amdgcn_wmma_f32_16x16x4_f32(false, a, false, b,
                                                    (short)0, acc, false, false);
      }
      const float bv = bhh[gcol];
      const int r0 = half * 8;
#pragma unroll
      for (int i = 0; i < 8; ++i) {
        const int r = r0 + i;
        if (r < B_) gh[r * (3 * O_) + gcol] = acc[i] + bv;
      }
    }
    __syncthreads();
    // --- gates -----------------------------------------------------------
    for (int e2 = tid; e2 < B_ * O_; e2 += 512) {
      const int b = e2 >> 8;
      const int j = e2 & (O_ - 1);
      const float* __restrict__ xrow = xp + ((size_t)b * L_ + t) * (3 * O_);
      const float xr = xrow[j];
      const float xz = xrow[O_ + j];
      const float xn = xrow[2 * O_ + j];
      const float hr = gh[b * (3 * O_) + j];
      const float hz = gh[b * (3 * O_) + O_ + j];
      const float hn = gh[b * (3 * O_) + 2 * O_ + j];
      const float r = 1.f / (1.f + expf(-(xr + hr)));
      const float z = 1.f / (1.f + expf(-(xz + hz)));
      const float n = tanhf(xn + r * hn);
      h[b * O_ + j] = (1.f - z) * n + z * h[b * O_ + j];
    }
    __syncthreads();
  }
  for (int e2 = tid; e2 < B_ * O_; e2 += 512) inv[e2] = h[e2];
}

// ---------------------------------------------------------------------------
// logits[b,a] = dot(act_proj[b,a,:], inv[b,:]) + dot(act_emb[b,a,:], abw) +
//               abb + max(log(mask), -FLT_MAX)
// ---------------------------------------------------------------------------
__global__ void logits_kernel(const float* __restrict__ aproj,
                              const float* __restrict__ aemb,
                              const float* __restrict__ inv,
                              const float* __restrict__ abw,
                              const float* __restrict__ abb,
                              const float* __restrict__ mask,
                              float* __restrict__ out) {
  __shared__ float s1[D_];
  __shared__ float s2[D_];
  const int idx = blockIdx.x;                  // b*A + a
  const int b = idx >> 7;                      // A_ == 128
  const int d = threadIdx.x;
  const size_t row = (size_t)idx * D_;
  s1[d] = aproj[row + d] * inv[b * O_ + d];
  s2[d] = aemb[row + d] * abw[d];
  __syncthreads();
  for (int s = D_ / 2; s > 0; s >>= 1) {
    if (d < s) { s1[d] += s1[d + s]; s2[d] += s2[d + s]; }
    __syncthreads();
  }
  if (d == 0) {
    const float fmin = -3.4028234663852886e38f;
    out[idx] = s1[0] + (s2[0] + abb[0]) + fmaxf(logf(mask[idx]), fmin);
  }
}

// ---------------------------------------------------------------------------
// Host-side launch
// ---------------------------------------------------------------------------
extern "C" void kernel_launch(void* const* d_in, const int* in_sizes, int n_in,
                              void* d_out, int out_size, void* d_ws, size_t ws_size,
                              hipStream_t stream) {
  const float* emb_table  = (const float*)d_in[0];   // [V,D]
  const float* gcn_w      = (const float*)d_in[1];   // [D,D]
  const float* gcn_b      = (const float*)d_in[2];   // [D]
  const float* proj_w     = (const float*)d_in[3];   // [D,H]
  const float* proj_b     = (const float*)d_in[4];   // [H]
  const float* gru_w_ih   = (const float*)d_in[5];   // [3O,H]
  const float* gru_w_hh   = (const float*)d_in[6];   // [3O,O]
  const float* gru_b_ih   = (const float*)d_in[7];   // [3O]
  const float* gru_b_hh   = (const float*)d_in[8];   // [3O]
  const float* act_proj_w = (const float*)d_in[9];   // [D,O]
  const float* act_proj_b = (const float*)d_in[10];  // [O]
  const float* act_bias_w = (const float*)d_in[11];  // [D,1]
  const float* act_bias_b = (const float*)d_in[12];  // [1]
  const float* action_mask= (const float*)d_in[13];  // [B,A]
  const int*   nn_seq     = (const int*)d_in[14];    // [B,L]
  const int*   all_actions= (const int*)d_in[15];    // [B,A]
  const int*   graph_x_tok= (const int*)d_in[16];    // [B,N]
  const int*   edge_index = (const int*)d_in[17];    // [B,2,E]
  float* out = (float*)d_out;                        // [B,A]

  // workspace layout (floats)
  float* ws = (float*)d_ws;
  const size_t SZ_TXW  = 1008 * D_;            // padded table_xw
  const size_t SZ_DEG  = (size_t)B_ * N_;
  const size_t SZ_AGG  = (size_t)B_ * N_ * D_;
  const size_t SZ_IEMB = (size_t)B_ * L_ * D_;
  const size_t SZ_IH   = (size_t)B_ * L_ * H_;
  const size_t SZ_XP   = (size_t)B_ * L_ * 3 * O_;
  const size_t SZ_INV  = (size_t)B_ * O_;
  const size_t SZ_AEMB = (size_t)B_ * A_ * D_;
  float* txw   = ws;                   ws += SZ_TXW;
  float* deg   = ws;                   ws += SZ_DEG;
  float* dinv  = ws;                   ws += SZ_DEG;
  float* agg   = ws;                   ws += SZ_AGG;   // becomes graph_repr
  float* iemb  = ws;                   ws += SZ_IEMB;
  float* ih    = ws;                   ws += SZ_IH;
  float* xp    = ws;                   ws += SZ_XP;
  float* invb  = ws;                   ws += SZ_INV;
  float* aemb  = ws;                   ws += SZ_AEMB;
  float* aproj = ws;                   ws += SZ_AEMB;

  // 0) init scratch accumulators
  fill_f32<<<dim3((unsigned)((SZ_AGG + 255) / 256)), dim3(256), 0, stream>>>(agg, SZ_AGG, 0.f);
  fill_f32<<<dim3((unsigned)((SZ_DEG + 255) / 256)), dim3(256), 0, stream>>>(deg, SZ_DEG, 1.f);

  // 1) table_xw = emb_table @ gcn_w          [1000,256]x[256,256]
  {
    int mt = (V_ + 15) / 16, nt = D_ / 16;
    gemm_wmma_f32<0, 0><<<dim3((mt * nt + 3) / 4), dim3(128), 0, stream>>>(
        emb_table, gcn_w, txw, nullptr, V_, D_, D_);
  }

  // 2) degrees + 1/sqrt
  edge_deg<<<dim3((B_ * E_ + 255) / 256), dim3(256), 0, stream>>>(edge_index, deg);
  deg_to_dinv<<<dim3((B_ * N_ + 255) / 256), dim3(256), 0, stream>>>(deg, dinv);

  // 3) edge scatter-add, then self-loop + bias + relu (in place -> graph_repr)
  edge_agg<<<dim3(B_ * E_), dim3(D_), 0, stream>>>(edge_index, graph_x_tok, txw, dinv, agg);
  graph_finish<<<dim3(B_ * N_), dim3(D_), 0, stream>>>(graph_x_tok, txw, dinv, gcn_b, agg);

  // 4) mixed embeddings
  mixed_embed<<<dim3(B_ * L_), dim3(D_), 0, stream>>>(nn_seq, emb_table, agg, iemb, L_);
  mixed_embed<<<dim3(B_ * A_), dim3(D_), 0, stream>>>(all_actions, emb_table, agg, aemb, A_);

  // 5) inv_h = relu(inv_emb @ proj_w + proj_b)      [512,256]x[256,512]
  {
    int mt = (B_ * L_) / 16, nt = H_ / 16;
    gemm_wmma_f32<0, 1><<<dim3((mt * nt + 3) / 4), dim3(128), 0, stream>>>(
        iemb, proj_w, ih, proj_b, B_ * L_, D_, H_);
  }

  // 6) xp = inv_h @ W_ih^T + b_ih                   [512,512]x[512,768]
  {
    int mt = (B_ * L_) / 16, nt = (3 * O_) / 16;
    gemm_wmma_f32<1, 0><<<dim3((mt * nt + 3) / 4), dim3(128), 0, stream>>>(
        ih, gru_w_ih, xp, gru_b_ih, B_ * L_, H_, 3 * O_);
  }

  // 7) act_proj = act_emb @ act_proj_w + act_proj_b [1024,256]x[256,256]
  {
    int mt = (B_ * A_) / 16, nt = O_ / 16;
    gemm_wmma_f32<0, 0><<<dim3((mt * nt + 3) / 4), dim3(128), 0, stream>>>(
        aemb, act_proj_w, aproj, act_proj_b, B_ * A_, D_, O_);
  }

  // 8) GRU recurrence (persistent, h in LDS)
  gru_kernel<<<dim3(1), dim3(512), 0, stream>>>(xp, gru_w_hh, gru_b_hh, invb);

  // 9) logits
  logits_kernel<<<dim3(B_ * A_), dim3(D_), 0, stream>>>(
      aproj, aemb, invb, act_bias_w, act_bias_b, action_mask, out);
}